// GCNLayer_51049981280832
// MI455X (gfx1250) — compile-verified
//
#include <hip/hip_runtime.h>

typedef __attribute__((ext_vector_type(2))) float v2f;
typedef __attribute__((ext_vector_type(8))) float v8f;

#define NODES    100000
#define NEDGES   1600000
#define DIM      128
#define KEEP_INV (1.0f / 0.9f)

// ---------------------------------------------------------------------------
// 1) zero the accumulation buffer (d_out) and the degree buffer (in ws)
// ---------------------------------------------------------------------------
__global__ void zero2_kernel(float* __restrict__ a, long long na,
                             float* __restrict__ b, long long nb) {
  long long i      = (long long)blockIdx.x * blockDim.x + threadIdx.x;
  long long stride = (long long)gridDim.x * blockDim.x;
  for (long long j = i; j < na; j += stride) a[j] = 0.0f;
  for (long long j = i; j < nb; j += stride) b[j] = 0.0f;
}

// ---------------------------------------------------------------------------
// 2) degree of each destination node (self-loop folded in later as +1)
// ---------------------------------------------------------------------------
__global__ void degree_kernel(const long long* __restrict__ ei,
                              float* __restrict__ deg, int nedges) {
  int e = blockIdx.x * blockDim.x + threadIdx.x;
  if (e >= nedges) return;
  long long d = ei[(long long)nedges + e];   // dst row of edge_index
  atomicAdd(&deg[d], 1.0f);
}

// ---------------------------------------------------------------------------
// 3) dinv[i] = rsqrt(deg[i] + 1)   (in place; +1 is the self loop)
// ---------------------------------------------------------------------------
__global__ void dinv_kernel(float* __restrict__ deg_dinv, int n) {
  int i = blockIdx.x * blockDim.x + threadIdx.x;
  if (i < n) deg_dinv[i] = rsqrtf(deg_dinv[i] + 1.0f);
}

// ---------------------------------------------------------------------------
// 4) h = x @ W with exact-fp32 WMMA (V_WMMA_F32_16X16X4_F32).
//    One wave -> one 16-row block, 8 accumulators cover all 128 cols.
//    NODES is an exact multiple of 16 (6250 tiles) so no ragged tile.
// ---------------------------------------------------------------------------
__global__ void __launch_bounds__(128)
gemm_wmma_f32(const float* __restrict__ x, const float* __restrict__ W,
              float* __restrict__ h, int nrows) {
  const int lane  = threadIdx.x & 31;
  const int wave  = blockIdx.x * (blockDim.x >> 5) + (threadIdx.x >> 5);
  const int tileM = wave << 4;
  if (tileM >= nrows) return;            // uniform per-wave guard, EXEC stays all-1s

  const int half = lane >> 4;            // 0: K=0,1   1: K=2,3
  const int l16  = lane & 15;            // A: row-in-tile, B: col-in-tile

  v8f zero = {0.f, 0.f, 0.f, 0.f, 0.f, 0.f, 0.f, 0.f};
  v8f acc[8];
#pragma unroll
  for (int n = 0; n < 8; ++n) acc[n] = zero;

  // A pointer: row tileM+l16, base K offset half*2
  const float* __restrict__ arow = x + (size_t)(tileM + l16) * DIM + (half << 1);

#pragma unroll 4
  for (int k = 0; k < DIM; k += 4) {
    v2f a;
    a.x = arow[k + 0];
    a.y = arow[k + 1];
    const float* __restrict__ bcol = W + (size_t)(k + (half << 1)) * DIM + l16;
#pragma unroll
    for (int n = 0; n < 8; ++n) {
      v2f b;
      b.x = bcol[n * 16];                // W[k+half*2+0][n*16+l16]
      b.y = bcol[n * 16 + DIM];          // W[k+half*2+1][n*16+l16]
      acc[n] = __builtin_amdgcn_wmma_f32_16x16x4_f32(
          /*neg_a=*/false, a, /*neg_b=*/false, b,
          /*c_mod=*/(short)0, acc[n], /*reuse_a=*/false, /*reuse_b=*/false);
    }
  }

  // D layout: VGPR j -> row tileM + j + 8*half, col = n*16 + l16
#pragma unroll
  for (int n = 0; n < 8; ++n) {
#pragma unroll
    for (int j = 0; j < 8; ++j) {
      int row = tileM + j + (half << 3);
      h[(size_t)row * DIM + n * 16 + l16] = acc[n][j];
    }
  }
}

// ---------------------------------------------------------------------------
// 5) edge scatter: out[dst] += dinv[src]*dinv[dst] * h[src]
//    One wave per edge; float4 per lane (coalesced 512B row, L2-resident).
// ---------------------------------------------------------------------------
__global__ void __launch_bounds__(256)
scatter_kernel(const float* __restrict__ h, const long long* __restrict__ ei,
               const float* __restrict__ dinv, float* __restrict__ out,
               int nedges) {
  const int lane = threadIdx.x & 31;
  const long long wave =
      (long long)blockIdx.x * (blockDim.x >> 5) + (threadIdx.x >> 5);
  if (wave >= nedges) return;

  const long long s = ei[wave];                      // src
  const long long d = ei[(long long)nedges + wave];  // dst
  const float norm  = dinv[s] * dinv[d];

  const float4 v = ((const float4*)(h + s * DIM))[lane];
  float* od = out + d * DIM + lane * 4;
  atomicAdd(od + 0, norm * v.x);
  atomicAdd(od + 1, norm * v.y);
  atomicAdd(od + 2, norm * v.z);
  atomicAdd(od + 3, norm * v.w);
}

// ---------------------------------------------------------------------------
// 6) self-loop + bias + ReLU + deterministic inverted dropout
// ---------------------------------------------------------------------------
__global__ void finalize_kernel(float* __restrict__ out, const float* __restrict__ h,
                                const float* __restrict__ dinv,
                                const float* __restrict__ bias,
                                const float* __restrict__ drop_u, long long total) {
  long long i = (long long)blockIdx.x * blockDim.x + threadIdx.x;
  if (i >= total) return;
  int c         = (int)(i & (DIM - 1));
  long long row = i >> 7;                 // DIM == 128
  float di  = dinv[row];
  float val = out[i] + di * di * h[i] + bias[c];
  val = fmaxf(val, 0.0f);
  val = (drop_u[i] >= 0.1f) ? val * KEEP_INV : 0.0f;
  out[i] = val;
}

// ---------------------------------------------------------------------------
extern "C" void kernel_launch(void* const* d_in, const int* in_sizes, int n_in,
                              void* d_out, int out_size, void* d_ws, size_t ws_size,
                              hipStream_t stream) {
  const float*     x      = (const float*)d_in[0];
  const long long* ei     = (const long long*)d_in[1];   // int64 [2, E]
  const float*     W      = (const float*)d_in[2];
  const float*     bias   = (const float*)d_in[3];
  const float*     drop_u = (const float*)d_in[4];
  float*           out    = (float*)d_out;

  // workspace layout: h [NODES*DIM] | deg/dinv [NODES]  (~51.6 MB)
  float* h    = (float*)d_ws;
  float* dinv = h + (size_t)NODES * DIM;

  const long long total = (long long)NODES * DIM;

  zero2_kernel<<<2048, 256, 0, stream>>>(out, total, dinv, (long long)NODES);
  degree_kernel<<<(NEDGES + 255) / 256, 256, 0, stream>>>(ei, dinv, NEDGES);
  dinv_kernel<<<(NODES + 255) / 256, 256, 0, stream>>>(dinv, NODES);

  // 6250 row-tiles, 4 waves per 128-thread block
  gemm_wmma_f32<<<(NODES / 16 + 3) / 4, 128, 0, stream>>>(x, W, h, NODES);

  // one wave per edge, 8 waves per block
  scatter_kernel<<<(NEDGES + 7) / 8, 256, 0, stream>>>(h, ei, dinv, out, NEDGES);

  finalize_kernel<<<(unsigned)((total + 255) / 256), 256, 0, stream>>>(
      out, h, dinv, bias, drop_u, total);
}